// GraphSage_19911468384623
// MI455X (gfx1250) — compile-verified
//
#include <hip/hip_runtime.h>

typedef __attribute__((ext_vector_type(2))) float v2f;
typedef __attribute__((ext_vector_type(8))) float v8f;

#define NNODES 50000
#define NEDGES 800000
#define NFEAT  128   // == N_HID, K dimension everywhere

// ---------------------------------------------------------------------------
// Utility: zero a float buffer
// ---------------------------------------------------------------------------
__global__ void zero_f32(float* __restrict__ p, long long n) {
    long long i = (long long)blockIdx.x * blockDim.x + threadIdx.x;
    if (i < n) p[i] = 0.0f;
}

// ---------------------------------------------------------------------------
// Degree: cnt[dst[e]] += 1  (fp32 atomics, L2-resident)
// ---------------------------------------------------------------------------
__global__ void degree_kernel(const long long* __restrict__ dst,
                              float* __restrict__ cnt, int E) {
    int e = blockIdx.x * blockDim.x + threadIdx.x;
    if (e < E) atomicAdd(&cnt[(int)dst[e]], 1.0f);
}

// ---------------------------------------------------------------------------
// Scatter-add: sums[dst] += feat[src].  One wave per edge, lane c handles
// float4 chunk c -> coalesced 512B row gather + contiguous atomic burst.
// ---------------------------------------------------------------------------
__global__ void scatter_kernel(const float* __restrict__ feat,
                               const long long* __restrict__ src,
                               const long long* __restrict__ dst,
                               float* __restrict__ sums, int E) {
    int t = blockIdx.x * blockDim.x + threadIdx.x;
    int e = t >> 5;
    if (e >= E) return;
    int c = t & 31;                       // 32 float4 chunks = 128 floats
    int s = (int)src[e];
    int d = (int)dst[e];
    const float4* row = (const float4*)(feat + (size_t)s * NFEAT);
    float4 v = row[c];
    float* p = sums + (size_t)d * NFEAT + c * 4;
    atomicAdd(p + 0, v.x);
    atomicAdd(p + 1, v.y);
    atomicAdd(p + 2, v.z);
    atomicAdd(p + 3, v.w);
}

// ---------------------------------------------------------------------------
// Mean finalize: sums[i] /= max(cnt[i/128], 1)
// ---------------------------------------------------------------------------
__global__ void mean_kernel(float* __restrict__ sums,
                            const float* __restrict__ cnt, long long n) {
    long long i = (long long)blockIdx.x * blockDim.x + threadIdx.x;
    if (i < n) sums[i] = sums[i] / fmaxf(cnt[i >> 7], 1.0f);
}

// ---------------------------------------------------------------------------
// Fused dual GEMM via fp32 WMMA:  D = act( A1@W1 + A2@W2 + bias )
// K = 128. One wave per 16-row strip: it computes ALL column tiles, holding
// TILES_N v8f accumulators so every A-fragment is reused TILES_N times.
// Weights staged once per block into LDS (zero-padded columns).
// A-frag (16x4 f32, 2 VGPRs): lane<16 -> M=lane, K={k,k+1}; lane>=16 -> K={k+2,k+3}
// B-frag: lane holds column (lane&15); VGPR0 row k+2*half, VGPR1 row k+2*half+1
// C: VGPR j = row (16*tileM + j + 8*half), col (16*t + (lane&15))
// ---------------------------------------------------------------------------
template<int NCOLS, bool RELU>
__global__ __launch_bounds__(256)
void sage_gemm_kernel(const float* __restrict__ A1, const float* __restrict__ W1,
                      const float* __restrict__ A2, const float* __restrict__ W2,
                      const float* __restrict__ bias, float* __restrict__ D) {
    constexpr int TILES_N = (NCOLS + 15) / 16;
    constexpr int NPAD    = TILES_N * 16;

    __shared__ float w1s[NFEAT * NPAD];
    __shared__ float w2s[NFEAT * NPAD];

    // Cooperative stage of both weight matrices into LDS (zero-pad columns).
    for (int i = threadIdx.x; i < NFEAT * NPAD; i += blockDim.x) {
        int k = i / NPAD, n = i % NPAD;
        float a = 0.0f, b = 0.0f;
        if (n < NCOLS) { a = W1[k * NCOLS + n]; b = W2[k * NCOLS + n]; }
        w1s[i] = a;
        w2s[i] = b;
    }
    __syncthreads();

    int wave  = (blockIdx.x * blockDim.x + threadIdx.x) >> 5;
    int lane  = threadIdx.x & 31;
    int tileM = wave;                    // one wave per 16-row strip
    if (tileM >= NNODES / 16) return;    // whole-wave exit after barrier

    int half  = lane >> 4;
    int nlane = lane & 15;
    int mrow  = tileM * 16 + nlane;      // A row this lane feeds

    const float* a1p = A1 + (size_t)mrow * NFEAT;
    const float* a2p = A2 + (size_t)mrow * NFEAT;

    v8f acc[TILES_N];
#pragma unroll
    for (int t = 0; t < TILES_N; ++t) acc[t] = (v8f){};

#pragma unroll 2
    for (int k = 0; k < NFEAT; k += 4) {
        int ka = k + 2 * half;
        v2f a1, a2;
        a1.x = a1p[ka];
        a1.y = a1p[ka + 1];
        a2.x = a2p[ka];
        a2.y = a2p[ka + 1];
#pragma unroll
        for (int t = 0; t < TILES_N; ++t) {
            int col = t * 16 + nlane;
            v2f b1, b2;
            b1.x = w1s[ka * NPAD + col];
            b1.y = w1s[(ka + 1) * NPAD + col];
            acc[t] = __builtin_amdgcn_wmma_f32_16x16x4_f32(false, a1, false, b1,
                                                           (short)0, acc[t], false, false);
            b2.x = w2s[ka * NPAD + col];
            b2.y = w2s[(ka + 1) * NPAD + col];
            acc[t] = __builtin_amdgcn_wmma_f32_16x16x4_f32(false, a2, false, b2,
                                                           (short)0, acc[t], false, false);
        }
    }

#pragma unroll
    for (int t = 0; t < TILES_N; ++t) {
        int col = t * 16 + nlane;
        float bval = (col < NCOLS) ? bias[col] : 0.0f;
#pragma unroll
        for (int j = 0; j < 8; ++j) {
            float v = acc[t][j] + bval;
            if (RELU) v = fmaxf(v, 0.0f);
            int row = tileM * 16 + j + 8 * half;
            if (col < NCOLS) D[(size_t)row * NCOLS + col] = v;
        }
    }
}

// ---------------------------------------------------------------------------
// Launch
// ---------------------------------------------------------------------------
extern "C" void kernel_launch(void* const* d_in, const int* in_sizes, int n_in,
                              void* d_out, int out_size, void* d_ws, size_t ws_size,
                              hipStream_t stream) {
    const float*     x   = (const float*)d_in[0];
    const long long* ei  = (const long long*)d_in[1];   // int64 [2, E]
    const float*     Wl1 = (const float*)d_in[2];
    const float*     Wr1 = (const float*)d_in[3];
    const float*     b1  = (const float*)d_in[4];
    const float*     Wl2 = (const float*)d_in[5];
    const float*     Wr2 = (const float*)d_in[6];
    const float*     b2  = (const float*)d_in[7];
    float*           out = (float*)d_out;

    const long long* src = ei;
    const long long* dst = ei + NEDGES;

    // workspace layout
    float* sums = (float*)d_ws;                          // N*128
    float* h    = sums + (size_t)NNODES * NFEAT;         // N*128
    float* cnt  = h    + (size_t)NNODES * NFEAT;         // N

    const long long NF = (long long)NNODES * NFEAT;      // 6.4M
    const int ZB_NF  = (int)((NF + 255) / 256);          // 25000
    const int ZB_N   = (NNODES + 255) / 256;             // 196
    const int EB     = (NEDGES + 255) / 256;             // 3125
    const int SB     = (int)(((long long)NEDGES * 32 + 255) / 256); // 100000
    const int GB     = (NNODES / 16 + 7) / 8;            // 391 blocks, 8 waves each

    // ---- Layer 1 aggregation ----
    zero_f32<<<ZB_NF, 256, 0, stream>>>(sums, NF);
    zero_f32<<<ZB_N,  256, 0, stream>>>(cnt, NNODES);
    degree_kernel<<<EB, 256, 0, stream>>>(dst, cnt, NEDGES);
    scatter_kernel<<<SB, 256, 0, stream>>>(x, src, dst, sums, NEDGES);
    mean_kernel<<<ZB_NF, 256, 0, stream>>>(sums, cnt, NF);

    // ---- Layer 1 dual GEMM + ReLU:  h = relu(agg@Wl1 + x@Wr1 + b1) ----
    sage_gemm_kernel<128, true><<<GB, 256, 0, stream>>>(sums, Wl1, x, Wr1, b1, h);

    // ---- Layer 2 aggregation (reuse sums, cnt unchanged) ----
    zero_f32<<<ZB_NF, 256, 0, stream>>>(sums, NF);
    scatter_kernel<<<SB, 256, 0, stream>>>(h, src, dst, sums, NEDGES);
    mean_kernel<<<ZB_NF, 256, 0, stream>>>(sums, cnt, NF);

    // ---- Layer 2 dual GEMM:  out = agg2@Wl2 + h@Wr2 + b2 ----
    sage_gemm_kernel<40, false><<<GB, 256, 0, stream>>>(sums, Wl2, h, Wr2, b2, out);
}